// GCNLayer_78675210928244
// MI455X (gfx1250) — compile-verified
//
#include <hip/hip_runtime.h>
#include <hip/hip_bf16.h>
#include <stdint.h>

#define ALPHA_C      0.1f
#define BETA_C       0.9f            // 1 - alpha
#define DFEAT        128
#define EDGES_PER_BLK 256

#define GLOBAL_AS __attribute__((address_space(1)))
#define LDS_AS    __attribute__((address_space(3)))

// ---- CDNA5 async global->LDS staging (ASYNCcnt path), with safe fallback ----
__device__ __forceinline__ void async_g2l_b32(const int* g, int* l) {
#if __has_builtin(__builtin_amdgcn_global_load_async_to_lds_b32)
    // builtin expects non-const AS(1) pointer; C-style cast drops const +
    // performs the generic->AS1 / generic->AS3 addrspace casts
    __builtin_amdgcn_global_load_async_to_lds_b32(
        (GLOBAL_AS int*)g, (LDS_AS int*)l, /*offset=*/0, /*cpol=*/0);
#else
    *l = *g;  // synchronous fallback (still correct)
#endif
}

__device__ __forceinline__ void async_wait_all() {
#if __has_builtin(__builtin_amdgcn_s_wait_asynccnt)
    __builtin_amdgcn_s_wait_asynccnt(0);
#elif __has_builtin(__builtin_amdgcn_global_load_async_to_lds_b32)
    asm volatile("s_wait_asynccnt 0x0" ::: "memory");
#endif
}

// ---- Kernel 1: out = ALPHA * h   (float4 streaming, saturates HBM) ----
__global__ __launch_bounds__(256) void gcn_scale_init(const float* __restrict__ h,
                                                      float* __restrict__ out,
                                                      int n4) {
    int i = blockIdx.x * 256 + threadIdx.x;
    if (i < n4) {
        float4 v = reinterpret_cast<const float4*>(h)[i];
        float4 o;
        o.x = ALPHA_C * v.x; o.y = ALPHA_C * v.y;
        o.z = ALPHA_C * v.z; o.w = ALPHA_C * v.w;
        reinterpret_cast<float4*>(out)[i] = o;
    }
}

// ---- Kernel 2: per-edge  out[dst] += BETA * h[src]  (wave32 per edge) ----
__global__ __launch_bounds__(256) void gcn_scatter(const float* __restrict__ h,
                                                   const int* __restrict__ src,
                                                   const int* __restrict__ dst,
                                                   float* __restrict__ out,
                                                   int n_edges) {
    __shared__ int s_src[EDGES_PER_BLK];
    __shared__ int s_dst[EDGES_PER_BLK];

    const int base  = blockIdx.x * EDGES_PER_BLK;
    int count = n_edges - base;
    if (count > EDGES_PER_BLK) count = EDGES_PER_BLK;   // >0 by grid sizing

    // Stage this block's edge indices into LDS via async copy (ASYNCcnt).
    const int t  = threadIdx.x;
    const int ge = base + ((t < count) ? t : (count - 1));   // clamp tail lanes
    async_g2l_b32(src + ge, &s_src[t]);
    async_g2l_b32(dst + ge, &s_dst[t]);
    async_wait_all();
    __syncthreads();

    const int lane = threadIdx.x & 31;   // wave32
    const int wave = threadIdx.x >> 5;   // 8 waves / block
    const int col  = lane * 4;           // 32 lanes x float4 = 128 cols

    for (int e = wave; e < count; e += 8) {
        const int s = s_src[e];          // LDS broadcast (wave-uniform)
        const int d = s_dst[e];

        const float4 v =
            *reinterpret_cast<const float4*>(h + (size_t)s * DFEAT + col);
        float* op = out + (size_t)d * DFEAT + col;

        // no-return global_atomic_add_f32 (STOREcnt, fire-and-forget)
        (void)__hip_atomic_fetch_add(op + 0, BETA_C * v.x,
                                     __ATOMIC_RELAXED, __HIP_MEMORY_SCOPE_AGENT);
        (void)__hip_atomic_fetch_add(op + 1, BETA_C * v.y,
                                     __ATOMIC_RELAXED, __HIP_MEMORY_SCOPE_AGENT);
        (void)__hip_atomic_fetch_add(op + 2, BETA_C * v.z,
                                     __ATOMIC_RELAXED, __HIP_MEMORY_SCOPE_AGENT);
        (void)__hip_atomic_fetch_add(op + 3, BETA_C * v.w,
                                     __ATOMIC_RELAXED, __HIP_MEMORY_SCOPE_AGENT);
    }
}

extern "C" void kernel_launch(void* const* d_in, const int* in_sizes, int n_in,
                              void* d_out, int out_size, void* d_ws, size_t ws_size,
                              hipStream_t stream) {
    (void)n_in; (void)d_ws; (void)ws_size;

    const float* h   = (const float*)d_in[0];
    const int*   src = (const int*)d_in[1];
    const int*   dst = (const int*)d_in[2];
    float*       out = (float*)d_out;

    const int n_edges = in_sizes[1];
    const int n4      = out_size / 4;          // N*D / 4 float4s

    gcn_scale_init<<<(n4 + 255) / 256, 256, 0, stream>>>(h, out, n4);

    const int nblk = (n_edges + EDGES_PER_BLK - 1) / EDGES_PER_BLK;
    gcn_scatter<<<nblk, 256, 0, stream>>>(h, src, dst, out, n_edges);
}